// LbmBGKLayer_51281909514865
// MI455X (gfx1250) — compile-verified
//
#include <hip/hip_runtime.h>

// D2Q9 LBM step, CDNA5 (gfx1250, wave32).
//
// out_j = (fs@M0)_j + (w_j/tau)*[rho + (4.5*(e_j.J)^2 - 1.5*|J|^2)/rho]
// with rho = sum_k fs_k, J = sum_k fs_k*e_k. The runtime 9x9 matmul (fs@M0)
// AND the moments rho/Jx/Jy are fused into one 16x16 WMMA f32 tile:
//   A (16 sites x K, K=9 padded to 12) -> 3 chained v_wmma_f32_16x16x4_f32
//   B cols 0..8 = M0, col 9 = ones (rho), col 10 = e_x, col 11 = e_y.
// Problem is bandwidth bound (~302 MB/step vs 23.3 TB/s HBM; the 151 MB field
// is L2-resident in 192 MB L2). The 1/rho uses v_rcp_f32 + 1 Newton step
// instead of the IEEE divide sequence to keep VALU off the critical path.

#define NXg 2048
#define NYg 2048
#define TAUc 0.6f

typedef __attribute__((ext_vector_type(2))) float v2f;
typedef __attribute__((ext_vector_type(8))) float v8f;

// E table packed as 2-bit fields: (e+1) per k, k=0..8
// ex: 0,1,0,-1,0,1,-1,-1,1   ey: 0,0,1,0,-1,1,1,-1,-1
__device__ __forceinline__ int exk(int k) { return ((0x20919 >> (2 * k)) & 3) - 1; }
__device__ __forceinline__ int eyk(int k) { return ((0x02865 >> (2 * k)) & 3) - 1; }

__device__ __forceinline__ float lane_bcast(float v, int srcLane) {
    int r = __builtin_amdgcn_ds_bpermute(srcLane << 2, __float_as_int(v));
    return __int_as_float(r);
}

// v_rcp_f32 (~1 ulp) + one Newton-Raphson refinement
__device__ __forceinline__ float fast_rcp(float x) {
    float r = __builtin_amdgcn_rcpf(x);
    r = r * (2.0f - x * r);
    return r;
}

__global__ __launch_bounds__(256) void lbm_step_kernel(const float* __restrict__ fin,
                                                       const float* __restrict__ M0,
                                                       float* __restrict__ fout) {
    const int x    = blockIdx.x;
    const int lane = threadIdx.x & 31;
    const int wave = threadIdx.x >> 5;
    const int half = lane >> 4;     // 0: lanes 0-15, 1: lanes 16-31
    const int n    = lane & 15;     // A-row site (for A) / C column (for C)

    // periodic x-wrap row bases; streaming reads row (x - e_x)
    const int xm = (x == 0) ? (NXg - 1) : (x - 1);
    const int xp = (x == NXg - 1) ? 0 : (x + 1);
    const float* rowX0 = fin + (size_t)x  * NYg * 9;
    const float* rowXm = fin + (size_t)xm * NYg * 9;  // e_x = +1
    const float* rowXp = fin + (size_t)xp * NYg * 9;  // e_x = -1
    float* outRow = fout + (size_t)x * NYg * 9;

    // Per-lane K indices for the 3 k-steps (f32 A/B layout: lanes 0-15 hold
    // K = 4s+0, 4s+1 in VGPR0/1; lanes 16-31 hold K = 4s+2, 4s+3).
    int kk[6];
#pragma unroll
    for (int s = 0; s < 3; ++s) { kk[2 * s] = 4 * s + 2 * half; kk[2 * s + 1] = kk[2 * s] + 1; }

    // Build B once per wave lifetime: rows k=0..8 (pad to 12),
    // cols: 0-8 = M0, 9 = 1 (rho), 10 = e_x(k), 11 = e_y(k), 12-15 = 0.
    v2f B[3];
#pragma unroll
    for (int s = 0; s < 3; ++s) {
        float b0 = 0.f, b1 = 0.f;
        const int kA = kk[2 * s], kB = kk[2 * s + 1];
        if (kA < 9) {
            if (n < 9)        b0 = M0[kA * 9 + n];
            else if (n == 9)  b0 = 1.0f;
            else if (n == 10) b0 = (float)exk(kA);
            else if (n == 11) b0 = (float)eyk(kA);
        }
        if (kB < 9) {
            if (n < 9)        b1 = M0[kB * 9 + n];
            else if (n == 9)  b1 = 1.0f;
            else if (n == 10) b1 = (float)exk(kB);
            else if (n == 11) b1 = (float)eyk(kB);
        }
        B[s].x = b0;
        B[s].y = b1;
    }

    // Per-lane output-column constants (column j = n)
    const float w_n = (n == 0) ? (4.0f / 9.0f) : ((n < 5) ? (1.0f / 9.0f) : (1.0f / 36.0f));
    const float wt  = w_n / TAUc;
    const float enx = (n < 9) ? (float)exk(n) : 0.f;
    const float eny = (n < 9) ? (float)eyk(n) : 0.f;
    const bool  active = (n < 9);
    const int srcRho = 9 + 16 * half;   // C col 9 lives in lane 9 (rows 0-7) / 25 (rows 8-15)
    const int srcJx  = 10 + 16 * half;
    const int srcJy  = 11 + 16 * half;

    // Each wave covers 16 tiles of 16 sites along y: 8 waves * 16 * 16 = 2048 = NY.
    for (int t = 0; t < 16; ++t) {
        const int y0 = (wave * 16 + t) * 16;
        const int ys = y0 + n;  // this lane's site row in the A tile

        // Gather streamed populations into WMMA-A layout: fs[m][k] = fin[x-ekx][y-eky][k]
        v2f A[3];
#pragma unroll
        for (int s = 0; s < 3; ++s) {
            float a0 = 0.f, a1 = 0.f;
            const int kA = kk[2 * s], kB = kk[2 * s + 1];
            if (kA < 9) {
                int yy = ys - eyk(kA);
                yy += (yy < 0) ? NYg : 0;
                yy -= (yy >= NYg) ? NYg : 0;
                const float* rb = (exk(kA) == 1) ? rowXm : ((exk(kA) == -1) ? rowXp : rowX0);
                a0 = rb[yy * 9 + kA];
            }
            if (kB < 9) {
                int yy = ys - eyk(kB);
                yy += (yy < 0) ? NYg : 0;
                yy -= (yy >= NYg) ? NYg : 0;
                const float* rb = (exk(kB) == 1) ? rowXm : ((exk(kB) == -1) ? rowXp : rowX0);
                a1 = rb[yy * 9 + kB];
            }
            A[s].x = a0;
            A[s].y = a1;
        }

        // C = sum over 12 K of A*B : [m0 | rho | Jx | Jy | 0...]
        v8f c = {};
        c = __builtin_amdgcn_wmma_f32_16x16x4_f32(false, A[0], false, B[0], (short)0, c, false, false);
        c = __builtin_amdgcn_wmma_f32_16x16x4_f32(false, A[1], false, B[1], (short)0, c, false, false);
        c = __builtin_amdgcn_wmma_f32_16x16x4_f32(false, A[2], false, B[2], (short)0, c, false, false);

        // Finish physics per site row r (lane holds column n of rows r + 8*half)
        float* p = outRow + (size_t)(y0 + 8 * half) * 9 + n;
#pragma unroll
        for (int r = 0; r < 8; ++r) {
            const float rho  = lane_bcast(c[r], srcRho);
            const float jx   = lane_bcast(c[r], srcJx);
            const float jy   = lane_bcast(c[r], srcJy);
            const float rrho = fast_rcp(rho);
            const float ej   = enx * jx + eny * jy;
            const float nl   = 4.5f * ej * ej - 1.5f * (jx * jx + jy * jy);
            // val = m0 + wt*rho + wt*nl/rho
            const float val  = __builtin_fmaf(wt * nl, rrho, __builtin_fmaf(wt, rho, c[r]));
            if (active) *p = val;
            p += 9;
        }
    }
}

extern "C" void kernel_launch(void* const* d_in, const int* in_sizes, int n_in,
                              void* d_out, int out_size, void* d_ws, size_t ws_size,
                              hipStream_t stream) {
    const float* f  = (const float*)d_in[0];   // (2048, 2048, 9) f32
    const float* M0 = (const float*)d_in[1];   // (9, 9) f32
    float* out = (float*)d_out;
    float* ws  = (float*)d_ws;                 // needs NX*NY*9*4 = ~151 MB

    dim3 grid(NXg);
    dim3 block(256);
    // 4 LBM steps; grid-wide sync between steps via separate launches on `stream`.
    lbm_step_kernel<<<grid, block, 0, stream>>>(f,   M0, ws);
    lbm_step_kernel<<<grid, block, 0, stream>>>(ws,  M0, out);
    lbm_step_kernel<<<grid, block, 0, stream>>>(out, M0, ws);
    lbm_step_kernel<<<grid, block, 0, stream>>>(ws,  M0, out);
}